// ExpandedGeodesicDist_45827301048583
// MI455X (gfx1250) — compile-verified
//
#include <hip/hip_runtime.h>
#include <stdint.h>

// ---------------------------------------------------------------------------
// ExpandedGeodesicDist on MI455X (gfx1250)
//   dist^2(row) accumulated via V_WMMA_F32_16X16X4_F32 diagonal of (A * A^T),
//   A = 16x4 chunk of (data - x).  Data streamed HBM -> LDS with
//   global_load_async_to_lds_b128, double buffered per wave (ASYNCcnt).
//   HBM-bound: 1.024 GB / 23.3 TB/s ~= 44 us floor.
// ---------------------------------------------------------------------------

typedef float v2f __attribute__((ext_vector_type(2)));
typedef float v8f __attribute__((ext_vector_type(8)));

#define N_DATA    2000000
#define DIM       128
#define TILE_ROWS 16
#define NTILES    (N_DATA / TILE_ROWS)   // 125000 exact
#define ROWF      132                    // padded floats per LDS row (bank-conflict free)
#define TILEF     (TILE_ROWS * ROWF)     // 2112 floats per buffer
#define TILE_B    (TILEF * 4)            // 8448 bytes (16B multiple)
#define WPB       2                      // waves per block (64 threads)
#define BLOCKS    2048
#define NWAVES    (BLOCKS * WPB)         // 4096
#define TOPK      10
#define F_INF     __builtin_inff()

// ----- CDNA5 async global -> LDS copy (ASYNCcnt), per CDNA5 ISA ----
__device__ __forceinline__ void async_copy_b128(const float* g, uint32_t lds_byte) {
  asm volatile("global_load_async_to_lds_b128 %0, %1, off"
               :: "v"(lds_byte), "v"(g)
               : "memory");
}
__device__ __forceinline__ void wait_async_le16() {
  asm volatile("s_wait_asynccnt 0x10" ::: "memory");
}
__device__ __forceinline__ void wait_async_0() {
  asm volatile("s_wait_asynccnt 0x0" ::: "memory");
}

__device__ __forceinline__ float wave_min32(float v) {
#pragma unroll
  for (int off = 16; off >= 1; off >>= 1)
    v = fminf(v, __shfl_xor(v, off, 32));
  return v;
}

// ---------------------------------------------------------------------------
__global__ void __launch_bounds__(64)
knn_topk_kernel(const float* __restrict__ data,
                const float* __restrict__ x,
                float* __restrict__ cand) {
  __shared__ __align__(16) float smem[WPB * 2 * TILEF + DIM];  // 34.3 KB

  const int tid    = threadIdx.x;
  const int lane   = tid & 31;
  const int w      = tid >> 5;
  const int waveId = blockIdx.x * WPB + w;

  // stage query x into LDS once per block
  float* sx = smem + WPB * 2 * TILEF;
  sx[tid]      = x[tid];
  sx[tid + 64] = x[tid + 64];
  __syncthreads();

  float*         wbuf  = smem + w * 2 * TILEF;          // this wave's 2 buffers
  const uint32_t wbyte = (uint32_t)(uintptr_t)wbuf;     // raw LDS byte address

  const int mrow = lane & 15;          // row within 16-row tile (WMMA M)
  const int koff = (lane >> 4) << 1;   // K sub-offset per A-matrix layout

  float t[TOPK];
#pragma unroll
  for (int i = 0; i < TOPK; ++i) t[i] = F_INF;

  int tile = waveId;
  int buf  = 0;

  if (tile < NTILES) {  // prime buffer 0
    const float* g = data + (size_t)tile * (TILE_ROWS * DIM) + lane * 4;
    const uint32_t l = wbyte + (uint32_t)lane * 16u;
#pragma unroll
    for (int r = 0; r < TILE_ROWS; ++r)
      async_copy_b128(g + r * DIM, l + r * (ROWF * 4));
  }

  while (tile < NTILES) {
    const int next = tile + NWAVES;
    if (next < NTILES) {  // prefetch next tile into other buffer
      const float* g = data + (size_t)next * (TILE_ROWS * DIM) + lane * 4;
      const uint32_t l = wbyte + (uint32_t)(buf ^ 1) * (uint32_t)TILE_B +
                         (uint32_t)lane * 16u;
#pragma unroll
      for (int r = 0; r < TILE_ROWS; ++r)
        async_copy_b128(g + r * DIM, l + r * (ROWF * 4));
      wait_async_le16();  // in-order: current tile's 16 copies have landed
    } else {
      wait_async_0();
    }

    // C += A * A^T over K = 128 in chunks of 4 (fp32 WMMA).
    const float* lrow = wbuf + buf * TILEF + mrow * ROWF + koff;
    const float* lx   = sx + koff;
    v8f c = {};
#pragma unroll
    for (int kb = 0; kb < DIM; kb += 4) {
      v2f dv = *(const v2f*)(lrow + kb);   // ds_load_b64, conflict-free
      v2f xv = *(const v2f*)(lx + kb);     // broadcast
      v2f a  = dv - xv;
      c = __builtin_amdgcn_wmma_f32_16x16x4_f32(false, a, false, a,
                                                (short)0, c, false, false);
    }

    // Diagonal of C = squared distance per row; deliver row m to lane m.
    float d2 = F_INF;
#pragma unroll
    for (int r = 0; r < 8; ++r) {
      float lo = __shfl(c[r], r, 32);        // diag[m=r]     @ (lane r,    vgpr r)
      float hi = __shfl(c[r], r + 24, 32);   // diag[m=r+8]   @ (lane r+24, vgpr r)
      if (lane == r)     d2 = lo;
      if (lane == r + 8) d2 = hi;
    }

    // per-lane running top-10 (sorted ascending)
    if (d2 < t[TOPK - 1]) {
      float nv = d2;
#pragma unroll
      for (int i = 0; i < TOPK; ++i) {
        float lo = fminf(t[i], nv);
        nv = fmaxf(t[i], nv);
        t[i] = lo;
      }
    }

    buf ^= 1;
    tile = next;
  }

  // merge 32 sorted per-lane lists -> wave-uniform top-10
  float r[TOPK];
#pragma unroll
  for (int k = 0; k < TOPK; ++k) {
    const float m = wave_min32(t[0]);
    const unsigned msk = (unsigned)__ballot(t[0] == m);
    const int src = __ffs(msk) - 1;
    if (lane == src) {  // pop winner's head
#pragma unroll
      for (int i = 0; i < TOPK - 1; ++i) t[i] = t[i + 1];
      t[TOPK - 1] = F_INF;
    }
    r[k] = m;
  }
  if (lane < TOPK) {
    float v = r[0];
#pragma unroll
    for (int i = 1; i < TOPK; ++i)
      if (lane == i) v = r[i];
    cand[(size_t)waveId * TOPK + lane] = v;  // squared distances
  }
}

// ---------------------------------------------------------------------------
__global__ void __launch_bounds__(32)
finalize_kernel(const float* __restrict__ x, const float* __restrict__ y,
                const float* __restrict__ cand, float* __restrict__ out) {
  const int lane = threadIdx.x;

  // ||x - y|| over 128 dims: 4 per lane + wave reduce
  const float4 xv = ((const float4*)x)[lane];
  const float4 yv = ((const float4*)y)[lane];
  const float d0 = xv.x - yv.x, d1 = xv.y - yv.y;
  const float d2 = xv.z - yv.z, d3 = xv.w - yv.w;
  float s = d0 * d0 + d1 * d1 + d2 * d2 + d3 * d3;
#pragma unroll
  for (int off = 16; off >= 1; off >>= 1) s += __shfl_xor(s, off, 32);
  const float geo = sqrtf(s) * 0.5f;  // / MANIFOLD_SPEED

  // global top-10 over all wave candidates
  float t[TOPK];
#pragma unroll
  for (int i = 0; i < TOPK; ++i) t[i] = F_INF;
  for (int i = lane; i < NWAVES * TOPK; i += 32) {
    const float v = cand[i];
    if (v < t[TOPK - 1]) {
      float nv = v;
#pragma unroll
      for (int j = 0; j < TOPK; ++j) {
        float lo = fminf(t[j], nv);
        nv = fmaxf(t[j], nv);
        t[j] = lo;
      }
    }
  }

  float sum = 0.f;
#pragma unroll
  for (int k = 0; k < TOPK; ++k) {
    const float m = wave_min32(t[0]);
    const unsigned msk = (unsigned)__ballot(t[0] == m);
    const int src = __ffs(msk) - 1;
    if (lane == src) {
#pragma unroll
      for (int i = 0; i < TOPK - 1; ++i) t[i] = t[i + 1];
      t[TOPK - 1] = F_INF;
    }
    sum += sqrtf(m);
  }

  if (lane == 0) out[0] = geo + sum / 10.0f;
}

// ---------------------------------------------------------------------------
extern "C" void kernel_launch(void* const* d_in, const int* in_sizes, int n_in,
                              void* d_out, int out_size, void* d_ws, size_t ws_size,
                              hipStream_t stream) {
  (void)in_sizes; (void)n_in; (void)out_size; (void)ws_size;
  const float* x    = (const float*)d_in[0];
  const float* y    = (const float*)d_in[1];
  const float* data = (const float*)d_in[2];
  float* out  = (float*)d_out;
  float* cand = (float*)d_ws;  // NWAVES * TOPK * 4 = 160 KB

  knn_topk_kernel<<<BLOCKS, WPB * 32, 0, stream>>>(data, x, cand);
  finalize_kernel<<<1, 32, 0, stream>>>(x, y, cand, out);
}